// HGT_85134841741367
// MI455X (gfx1250) — compile-verified
//
#include <hip/hip_runtime.h>
#include <hip/hip_bf16.h>

// ---------------------------------------------------------------------------
// HGT forward for MI455X (gfx1250, wave32). fp32 WMMA 16x16x4 for all GEMMs.
// ---------------------------------------------------------------------------

typedef __attribute__((ext_vector_type(2))) float v2f;
typedef __attribute__((ext_vector_type(8))) float v8f;

#define N_NODES 10000
#define F_IN    256
#define HID     128
#define NHEAD   8
#define DH      16
#define NLAYER  2
#define E_EDGES 480000
#define P_PAIRS 200000

// ---------------------------------------------------------------------------
// GEMM: C[M,128] = act(A[M,K] @ B[K,128] + bias[128]).  M % 16 == 0, K % 4 == 0.
// One block = 16 rows of A staged in LDS; 8 waves each own a 16-col tile.
// fp32 WMMA 16x16x4, chained over K.
// ---------------------------------------------------------------------------
__global__ __launch_bounds__(256) void hgt_gemm_wmma_f32(
    const float* __restrict__ A, const float* __restrict__ B,
    const float* __restrict__ bias, float* __restrict__ C,
    int K, int act /*0=none,1=relu*/)
{
    extern __shared__ float As[];            // [16][K+2] padded (bank-conflict free)
    const int ldsA = K + 2;
    const int row0 = blockIdx.x * 16;

    // Stage the 16xK A tile (16*K is a multiple of 256 for K=128/256 -> no divergence)
    const int total = 16 * K;
    for (int i = threadIdx.x; i < total; i += 256) {
        int r = i / K, c = i - r * K;
        As[r * ldsA + c] = A[(size_t)(row0 + r) * K + c];
    }
    __syncthreads();

    const int wave = threadIdx.x >> 5;
    const int lane = threadIdx.x & 31;
    const int half = lane >> 4;              // 0: K pair {0,1}; 1: K pair {2,3}
    const int l16  = lane & 15;
    const int col  = wave * 16 + l16;

    v8f acc;
    const float bv = bias[col];
#pragma unroll
    for (int g = 0; g < 8; ++g) acc[g] = bv;

    for (int kk = 0; kk < K; kk += 4) {
        // A fragment: lane<16 -> A[m][kk+0..1], lane>=16 -> A[m][kk+2..3]
        const float* ap = &As[l16 * ldsA + kk + half * 2];
        v2f a = *(const v2f*)ap;             // 8B-aligned (even float index)
        // B fragment: lane<16 -> B[kk+0..1][col], lane>=16 -> B[kk+2..3][col]
        const float* bp = &B[(size_t)(kk + half * 2) * HID + col];
        v2f b; b.x = bp[0]; b.y = bp[HID];
        acc = __builtin_amdgcn_wmma_f32_16x16x4_f32(false, a, false, b,
                                                    (short)0, acc, false, false);
    }

#pragma unroll
    for (int g = 0; g < 8; ++g) {
        int r = row0 + g + half * 8;
        float v = acc[g];
        if (act == 1) v = v > 0.f ? v : 0.f;
        C[(size_t)r * HID + col] = v;
    }
}

// ---------------------------------------------------------------------------
// Per-head relation transform: kout[n,h,:] = kin[n,h,:] @ arel[h]  (16x16),
// vout likewise with mrel.  One wave per (16-node block, head). 8 WMMAs/wave.
// ---------------------------------------------------------------------------
__global__ __launch_bounds__(256) void hgt_rel_transform(
    const float* __restrict__ kin, const float* __restrict__ vin,
    const float* __restrict__ arel, const float* __restrict__ mrel,
    float* __restrict__ kout, float* __restrict__ vout)
{
    const int n0   = blockIdx.x * 16;
    const int h    = threadIdx.x >> 5;       // 8 waves = 8 heads
    const int lane = threadIdx.x & 31;
    const int half = lane >> 4;
    const int l16  = lane & 15;

    const float* ah = arel + h * (DH * DH);
    const float* mh = mrel + h * (DH * DH);

    v8f ka = {0.f,0.f,0.f,0.f,0.f,0.f,0.f,0.f};
    v8f va = {0.f,0.f,0.f,0.f,0.f,0.f,0.f,0.f};

#pragma unroll
    for (int kk = 0; kk < DH; kk += 4) {
        const float* kp = kin + (size_t)(n0 + l16) * HID + h * DH + kk + half * 2;
        const float* vp = vin + (size_t)(n0 + l16) * HID + h * DH + kk + half * 2;
        v2f a1; a1.x = kp[0]; a1.y = kp[1];
        v2f a2; a2.x = vp[0]; a2.y = vp[1];
        v2f b1; b1.x = ah[(kk + half * 2) * DH + l16]; b1.y = ah[(kk + half * 2 + 1) * DH + l16];
        v2f b2; b2.x = mh[(kk + half * 2) * DH + l16]; b2.y = mh[(kk + half * 2 + 1) * DH + l16];
        ka = __builtin_amdgcn_wmma_f32_16x16x4_f32(false, a1, false, b1, (short)0, ka, false, false);
        va = __builtin_amdgcn_wmma_f32_16x16x4_f32(false, a2, false, b2, (short)0, va, false, false);
    }

#pragma unroll
    for (int g = 0; g < 8; ++g) {
        int n = n0 + g + half * 8;
        kout[(size_t)n * HID + h * DH + l16] = ka[g];
        vout[(size_t)n * HID + h * DH + l16] = va[g];
    }
}

// ---------------------------------------------------------------------------
// Edge-attention kernels (segment softmax over dst, per head)
// ---------------------------------------------------------------------------
__device__ __forceinline__ void atomicMaxFloat(float* addr, float val) {
    // init pattern 0xFFFFFFFF; signed-max for >=0, unsigned-min for <0
    if (!(val < 0.f)) atomicMax((int*)addr, __float_as_int(val));
    else              atomicMin((unsigned int*)addr, __float_as_uint(val));
}

__global__ __launch_bounds__(256) void hgt_edge_logits(
    const float* __restrict__ q, const float* __restrict__ kt,
    const int* __restrict__ src, const int* __restrict__ dst,
    const float* __restrict__ prior, float* __restrict__ logits,
    float* __restrict__ mbuf, int E)
{
    int idx = blockIdx.x * 256 + threadIdx.x;
    if (idx >= E * NHEAD) return;
    int e = idx >> 3, h = idx & 7;
    int s = src[e], d = dst[e];
    const float4* qp = (const float4*)(q  + (size_t)d * HID + h * DH);
    const float4* kp = (const float4*)(kt + (size_t)s * HID + h * DH);
    float acc = 0.f;
#pragma unroll
    for (int i = 0; i < 4; ++i) {
        float4 x = qp[i], y = kp[i];
        acc += x.x * y.x + x.y * y.y + x.z * y.z + x.w * y.w;
    }
    float lg = acc * prior[h] * 0.25f;       // 1/sqrt(DH) = 0.25
    logits[idx] = lg;
    atomicMaxFloat(&mbuf[d * NHEAD + h], lg);
}

__global__ __launch_bounds__(256) void hgt_edge_expsum(
    float* __restrict__ logits, const float* __restrict__ mbuf,
    float* __restrict__ sbuf, const int* __restrict__ dst, int E)
{
    int idx = blockIdx.x * 256 + threadIdx.x;
    if (idx >= E * NHEAD) return;
    int e = idx >> 3, h = idx & 7;
    int d = dst[e];
    float ev = __expf(logits[idx] - mbuf[d * NHEAD + h]);
    logits[idx] = ev;
    atomicAdd(&sbuf[d * NHEAD + h], ev);
}

__global__ __launch_bounds__(256) void hgt_edge_aggregate(
    const float* __restrict__ logits, const float* __restrict__ sbuf,
    const float* __restrict__ vt, const int* __restrict__ src,
    const int* __restrict__ dst, float* __restrict__ agg, int E)
{
    int idx = blockIdx.x * 256 + threadIdx.x;
    if (idx >= E * 32) return;
    int e = idx >> 5;
    int part = idx & 31;
    int h = part >> 2;
    int c = (part & 3) * 4;
    int d = dst[e], s = src[e];
    float a = logits[e * NHEAD + h] / (sbuf[d * NHEAD + h] + 1e-16f);
    float4 v4 = *(const float4*)(vt + (size_t)s * HID + h * DH + c);
    float* op = agg + (size_t)d * HID + h * DH + c;
    atomicAdd(op + 0, a * v4.x);
    atomicAdd(op + 1, a * v4.y);
    atomicAdd(op + 2, a * v4.z);
    atomicAdd(op + 3, a * v4.w);
}

// ---------------------------------------------------------------------------
// Elementwise: GELU (tanh approx, JAX default) and sigmoid-gated skip update.
// ---------------------------------------------------------------------------
__global__ __launch_bounds__(256) void hgt_gelu_inplace(float* __restrict__ x, int n)
{
    int i = blockIdx.x * 256 + threadIdx.x;
    if (i >= n) return;
    float v = x[i];
    float t = tanhf(0.7978845608028654f * (v + 0.044715f * v * v * v));
    x[i] = 0.5f * v * (1.f + t);
}

__global__ __launch_bounds__(256) void hgt_skip_update(
    float* __restrict__ h, const float* __restrict__ o,
    const float* __restrict__ skip, int skip_idx,
    float* __restrict__ feat, int layer, int n)
{
    int i = blockIdx.x * 256 + threadIdx.x;
    if (i >= n) return;
    float sg = 1.f / (1.f + __expf(-skip[skip_idx]));
    float hv = sg * o[i] + (1.f - sg) * h[i];
    h[i] = hv;
    int node = i >> 7, c = i & 127;          // HID = 128
    feat[(size_t)node * (HID * NLAYER) + layer * HID + c] = hv;
}

// ---------------------------------------------------------------------------
// Final gathered dot: out[p] = dot(Em[i[p]], Ed[j[p]]), 256 dims, 8 lanes/pair.
// ---------------------------------------------------------------------------
__global__ __launch_bounds__(256) void hgt_pairdot(
    const float* __restrict__ Em, const float* __restrict__ Ed,
    const int* __restrict__ pi, const int* __restrict__ pj,
    float* __restrict__ out, int P)
{
    int idx = blockIdx.x * 256 + threadIdx.x;
    int p = idx >> 3, t = idx & 7;
    if (p >= P) return;
    const float4* a = (const float4*)(Em + (size_t)pi[p] * 256 + t * 32);
    const float4* b = (const float4*)(Ed + (size_t)pj[p] * 256 + t * 32);
    float acc = 0.f;
#pragma unroll
    for (int i = 0; i < 8; ++i) {
        float4 x = a[i], y = b[i];
        acc += x.x * y.x + x.y * y.y + x.z * y.z + x.w * y.w;
    }
    acc += __shfl_xor(acc, 1, 8);
    acc += __shfl_xor(acc, 2, 8);
    acc += __shfl_xor(acc, 4, 8);
    if (t == 0) out[p] = acc;
}

// ---------------------------------------------------------------------------
// Orchestration
// ---------------------------------------------------------------------------
extern "C" void kernel_launch(void* const* d_in, const int* in_sizes, int n_in,
                              void* d_out, int out_size, void* d_ws, size_t ws_size,
                              hipStream_t stream)
{
    const float* x1   = (const float*)d_in[0];
    const float* x2   = (const float*)d_in[1];
    const int*   ei12 = (const int*)  d_in[2];   // [2, E]: src(n1), dst(n2)
    const int*   ei21 = (const int*)  d_in[3];   // [2, E]: src(n2), dst(n1)
    const int*   pidx = (const int*)  d_in[4];   // [2, P]
    const float* Win1 = (const float*)d_in[5];
    const float* bin1 = (const float*)d_in[6];
    const float* Win2 = (const float*)d_in[7];
    const float* bin2 = (const float*)d_in[8];
    const float* Wk   = (const float*)d_in[9];
    const float* bk   = (const float*)d_in[10];
    const float* Wq   = (const float*)d_in[11];
    const float* bq   = (const float*)d_in[12];
    const float* Wv   = (const float*)d_in[13];
    const float* bv   = (const float*)d_in[14];
    const float* Wa   = (const float*)d_in[15];
    const float* ba   = (const float*)d_in[16];
    const float* skip = (const float*)d_in[17];
    const float* arel = (const float*)d_in[18];  // [L,2,H,16,16]
    const float* mrel = (const float*)d_in[19];
    const float* prior= (const float*)d_in[20];  // [L,2,H]

    const size_t NF = (size_t)N_NODES * HID;     // 1,280,000 floats
    float* ws = (float*)d_ws;
    float* h1  = ws;             float* h2  = h1  + NF;
    float* kb[2] = { h2 + NF,        h2 + 2*NF };
    float* qb[2] = { h2 + 3*NF,      h2 + 4*NF };
    float* vb[2] = { h2 + 5*NF,      h2 + 6*NF };
    float* kt  = h2 + 7*NF;      float* vt  = h2 + 8*NF;
    float* agg = h2 + 9*NF;      float* ob  = h2 + 10*NF;
    float* Em  = h2 + 11*NF;                       // N x 256
    float* Ed  = Em + (size_t)N_NODES * 256;
    float* logits = Ed + (size_t)N_NODES * 256;    // E x 8
    float* mbuf   = logits + (size_t)E_EDGES * NHEAD;
    float* sbuf   = mbuf + (size_t)N_NODES * NHEAD;

    float* hbuf[2] = { h1, h2 };
    float* feat[2] = { Em, Ed };

    const dim3 blk(256);
    const dim3 gM16(N_NODES / 16);                          // 625
    const int shm128 = 16 * (HID + 2) * sizeof(float);
    const int shm256 = 16 * (F_IN + 2) * sizeof(float);
    const int edgeH_blocks  = (E_EDGES * NHEAD + 255) / 256;
    const int edge32_blocks = (E_EDGES * 32 + 255) / 256;
    const int nf_blocks     = (int)((NF + 255) / 256);

    // Input projections with fused ReLU
    hipLaunchKernelGGL(hgt_gemm_wmma_f32, gM16, blk, shm256, stream,
                       x1, Win1, bin1, h1, F_IN, 1);
    hipLaunchKernelGGL(hgt_gemm_wmma_f32, gM16, blk, shm256, stream,
                       x2, Win2, bin2, h2, F_IN, 1);

    for (int l = 0; l < NLAYER; ++l) {
        // K/Q/V projections for both node types (from pre-update h)
        for (int t = 0; t < 2; ++t) {
            const size_t wo = (size_t)(l * 2 + t) * HID * HID;
            const size_t bo = (size_t)(l * 2 + t) * HID;
            hipLaunchKernelGGL(hgt_gemm_wmma_f32, gM16, blk, shm128, stream,
                               hbuf[t], Wk + wo, bk + bo, kb[t], HID, 0);
            hipLaunchKernelGGL(hgt_gemm_wmma_f32, gM16, blk, shm128, stream,
                               hbuf[t], Wq + wo, bq + bo, qb[t], HID, 0);
            hipLaunchKernelGGL(hgt_gemm_wmma_f32, gM16, blk, shm128, stream,
                               hbuf[t], Wv + wo, bv + bo, vb[t], HID, 0);
        }

        for (int et = 0; et < 2; ++et) {
            const int st = (et == 0) ? 0 : 1;     // source node type
            const int dt = 1 - et;                // destination node type
            const int* src = (et == 0) ? ei12 : ei21;
            const int* dst = src + E_EDGES;
            const size_t ro = (size_t)(l * 2 + et) * NHEAD * DH * DH;

            // k/v relation transforms (per-head 16x16 WMMA)
            hipLaunchKernelGGL(hgt_rel_transform, gM16, blk, 0, stream,
                               kb[st], vb[st], arel + ro, mrel + ro, kt, vt);

            hipMemsetAsync(mbuf, 0xFF, (size_t)N_NODES * NHEAD * sizeof(float), stream);
            hipMemsetAsync(sbuf, 0,    (size_t)N_NODES * NHEAD * sizeof(float), stream);
            hipMemsetAsync(agg,  0,    NF * sizeof(float), stream);

            hipLaunchKernelGGL(hgt_edge_logits, dim3(edgeH_blocks), blk, 0, stream,
                               qb[dt], kt, src, dst,
                               prior + (size_t)(l * 2 + et) * NHEAD,
                               logits, mbuf, E_EDGES);
            hipLaunchKernelGGL(hgt_edge_expsum, dim3(edgeH_blocks), blk, 0, stream,
                               logits, mbuf, sbuf, dst, E_EDGES);
            hipLaunchKernelGGL(hgt_edge_aggregate, dim3(edge32_blocks), blk, 0, stream,
                               logits, sbuf, vt, src, dst, agg, E_EDGES);

            // o = gelu(agg) @ Wa[l,dt] + ba[l,dt]
            hipLaunchKernelGGL(hgt_gelu_inplace, dim3(nf_blocks), blk, 0, stream,
                               agg, (int)NF);
            const size_t ao = (size_t)(l * 2 + dt) * HID * HID;
            hipLaunchKernelGGL(hgt_gemm_wmma_f32, gM16, blk, shm128, stream,
                               agg, Wa + ao, ba + (size_t)(l * 2 + dt) * HID, ob, HID, 0);

            // h[dt] = sig(skip)*o + (1-sig)*h[dt]; also write concat feature slice
            hipLaunchKernelGGL(hgt_skip_update, dim3(nf_blocks), blk, 0, stream,
                               hbuf[dt], ob, skip, l * 2 + dt, feat[dt], l, (int)NF);
        }
    }

    // preds[p] = dot(Em[i[p]], Ed[j[p]])
    const int pd_blocks = (P_PAIRS * 8 + 255) / 256;
    hipLaunchKernelGGL(hgt_pairdot, dim3(pd_blocks), blk, 0, stream,
                       Em, Ed, pidx, pidx + P_PAIRS, (float*)d_out, P_PAIRS);
}